// ArcFaceLoss_48017734369761
// MI455X (gfx1250) — compile-verified
//
#include <hip/hip_runtime.h>
#include <hip/hip_bf16.h>
#include <math.h>

// ---------------------------------------------------------------------------
// ArcFace loss, MI455X (gfx1250, wave32).
//   B=2048 rows, D=512 features, C=50000 classes, margin=0.3, scale=30.
// Strategy: single HBM pass over weight (the 102MB / 23.3TB/s ~ 4.4us roofline
// term). Each block caches a 128-column weight slice (pre-scaled by
// SCALE/||w_c||) in 264KB of LDS, streams all embedding rows (pre-scaled by
// 1/||e_b||) through V_WMMA_F32_16X16X4_F32, and emits per-(row, colblock)
// online-softmax partials. Two small reduction kernels finish the
// log-sum-exp merge and mean NLL.
// Round-2 refinements: dual WMMA accumulators (2 independent chains/wave)
// and register-staged prefetch of the next embedding tile so global loads
// overlap the WMMA loop.
// ---------------------------------------------------------------------------

typedef float v2f __attribute__((ext_vector_type(2)));
typedef float v8f __attribute__((ext_vector_type(8)));

#define BB     2048
#define DD     512
#define CC     50000
#define NT     128                    // columns per block
#define NBLK   391                    // ceil(50000 / 128)
#define DPAD   516                    // 512 + 4 floats: bank-conflict-free LDS rows
#define SMARG  9.0f                   // SCALE * MARGIN = 30 * 0.3
#define NEGBIG (-1.0e30f)
#define RTILES (BB / 16)              // 128 row tiles

// LDS budget for the main kernel (bytes):
//   sB: 128*516*4 = 264192, sA: 16*516*4 = 33024, red: 2*128*4 = 1024, slab: 64
#define SMEM_BYTES (NT * DPAD * 4 + 16 * DPAD * 4 + 256 * 4 + 64)

// --------------------------- row inverse-norms -----------------------------
__global__ __launch_bounds__(128) void arc_rownorm_kernel(
    const float* __restrict__ src, float* __restrict__ inv) {
  __shared__ float sc[4];
  const int row = blockIdx.x;
  const float4* p = (const float4*)(src + (size_t)row * DD);
  float s = 0.f;
  for (int q = threadIdx.x; q < DD / 4; q += 128) {
    float4 v = p[q];
    s += v.x * v.x + v.y * v.y + v.z * v.z + v.w * v.w;
  }
  for (int off = 16; off > 0; off >>= 1) s += __shfl_xor(s, off, 32);
  if ((threadIdx.x & 31) == 0) sc[threadIdx.x >> 5] = s;
  __syncthreads();
  if (threadIdx.x == 0) {
    float t = sc[0] + sc[1] + sc[2] + sc[3];
    inv[row] = 1.0f / fmaxf(sqrtf(t), 1e-12f);  // matches F.normalize eps
  }
}

// ------------------------------ main kernel --------------------------------
__global__ __launch_bounds__(256) void arc_main_kernel(
    const float* __restrict__ emb, const int* __restrict__ labels,
    const float* __restrict__ wgt, const float* __restrict__ inv_e,
    const float* __restrict__ inv_w, float* __restrict__ pmax,
    float* __restrict__ psum, float* __restrict__ loglab) {
  extern __shared__ __align__(16) float smem[];
  float* sB     = smem;                     // [NT][DPAD]  scaled weight slice
  float* sA     = sB + NT * DPAD;           // [16][DPAD]  scaled emb tile
  float* redmax = sA + 16 * DPAD;           // [8 waves][16 rows]
  float* redsum = redmax + 128;             // [8 waves][16 rows]
  int*   slab   = (int*)(redsum + 128);     // [16] labels of current row tile

  const int tid  = threadIdx.x;
  const int lane = tid & 31;
  const int wave = tid >> 5;
  const int blk  = blockIdx.x;
  const int c0   = blk * NT;

  // Stage weight slice into LDS, folding SCALE * (1/||w_c||). OOB cols -> 0.
  for (int idx = tid; idx < NT * (DD / 4); idx += 256) {
    const int j = idx >> 7;       // DD/4 == 128
    const int q = idx & 127;
    const int c = c0 + j;
    float4 v = make_float4(0.f, 0.f, 0.f, 0.f);
    float s = 0.f;
    if (c < CC) {
      v = ((const float4*)wgt)[(size_t)c * (DD / 4) + q];
      s = inv_w[c] * 30.0f;
    }
    v.x *= s; v.y *= s; v.z *= s; v.w *= s;
    *(float4*)(sB + j * DPAD + q * 4) = v;
  }

  // WMMA fragment addressing (wave32):
  //  A 16x4 f32: lane 0-15 -> M=lane holds K{0,1}; lane 16-31 -> M=lane-16, K{2,3}
  //  B 4x16 f32: lane 0-15 -> N=lane holds K{0,1}; lane 16-31 -> N=lane-16, K{2,3}
  //  C 16x16:    acc[g] on lane l is row m=(l>>4)*8+g, col n=l&15
  const int   n     = lane & 15;
  const int   koff  = (lane >> 4) * 2;
  const int   jbase = wave * 16;
  const int   mbase = (lane >> 4) * 8;
  const float* pBrow = sB + (jbase + n) * DPAD;
  const float* pArow = sA + (lane & 15) * DPAD;
  const int   cmy   = c0 + jbase + n;       // this lane's global column
  const bool  oob   = (cmy >= CC);
  const float4* emb4 = (const float4*)emb;

  // Register-staged prefetch of the embedding tile (8 float4 per thread).
  float4 ar[8];
#pragma unroll
  for (int i = 0; i < 8; ++i) {
    const int idx = tid + i * 256;           // 0..2047 over 16*128 float4s
    ar[i] = emb4[(size_t)(idx >> 7) * (DD / 4) + (idx & 127)];
  }

  for (int rt = 0; rt < RTILES; ++rt) {
    const int row0 = rt * 16;
    __syncthreads();  // previous iteration fully consumed sA / red buffers

    // Commit staged registers -> LDS with 1/||e_b|| folded in.
#pragma unroll
    for (int i = 0; i < 8; ++i) {
      const int idx = tid + i * 256;
      const int r = idx >> 7;
      const int q = idx & 127;
      const float se = inv_e[row0 + r];
      float4 v = ar[i];
      v.x *= se; v.y *= se; v.z *= se; v.w *= se;
      *(float4*)(sA + r * DPAD + q * 4) = v;
    }
    if (tid < 16) slab[tid] = labels[row0 + tid];
    __syncthreads();

    // Issue next tile's global loads now; they overlap the WMMA loop below
    // and are consumed only after the next barrier.
    if (rt + 1 < RTILES) {
      const int row0n = row0 + 16;
#pragma unroll
      for (int i = 0; i < 8; ++i) {
        const int idx = tid + i * 256;
        ar[i] = emb4[(size_t)(row0n + (idx >> 7)) * (DD / 4) + (idx & 127)];
      }
    }

    // Dual-accumulator WMMA loop: two independent dependency chains.
    v8f acc0 = {0.f, 0.f, 0.f, 0.f, 0.f, 0.f, 0.f, 0.f};
    v8f acc1 = {0.f, 0.f, 0.f, 0.f, 0.f, 0.f, 0.f, 0.f};
#pragma unroll 8
    for (int k0 = 0; k0 < DD; k0 += 8) {
      v2f a0 = *(const v2f*)(pArow + k0 + koff);
      v2f b0 = *(const v2f*)(pBrow + k0 + koff);
      v2f a1 = *(const v2f*)(pArow + k0 + 4 + koff);
      v2f b1 = *(const v2f*)(pBrow + k0 + 4 + koff);
      acc0 = __builtin_amdgcn_wmma_f32_16x16x4_f32(
          false, a0, false, b0, (short)0, acc0, false, false);
      acc1 = __builtin_amdgcn_wmma_f32_16x16x4_f32(
          false, a1, false, b1, (short)0, acc1, false, false);
    }

    // acc0+acc1 holds logits (SCALE * cos) for this wave's 16x16 tile.
#pragma unroll
    for (int g = 0; g < 8; ++g) {
      const int m = mbase + g;
      float logit = acc0[g] + acc1[g];
      if (oob) {
        logit = NEGBIG;                          // padded column
      } else if (slab[m] == cmy) {
        logit -= SMARG;                          // margin at the label
        loglab[row0 + m] = logit;                // unique writer grid-wide
      }
      float lm = logit;                          // max over the 16-lane row
      lm = fmaxf(lm, __shfl_xor(lm, 1, 32));
      lm = fmaxf(lm, __shfl_xor(lm, 2, 32));
      lm = fmaxf(lm, __shfl_xor(lm, 4, 32));
      lm = fmaxf(lm, __shfl_xor(lm, 8, 32));
      float e = expf(logit - lm);                // sum exp over the row
      e += __shfl_xor(e, 1, 32);
      e += __shfl_xor(e, 2, 32);
      e += __shfl_xor(e, 4, 32);
      e += __shfl_xor(e, 8, 32);
      if (n == 0) {                              // lanes 0 and 16 publish
        redmax[wave * 16 + m] = lm;
        redsum[wave * 16 + m] = e;
      }
    }
    __syncthreads();
    if (tid < 16) {  // merge 8 waves -> one partial per (row, block)
      float M = NEGBIG;
#pragma unroll
      for (int w = 0; w < 8; ++w) M = fmaxf(M, redmax[w * 16 + tid]);
      float S = 0.f;
#pragma unroll
      for (int w = 0; w < 8; ++w)
        S += redsum[w * 16 + tid] * expf(redmax[w * 16 + tid] - M);
      pmax[(size_t)(row0 + tid) * NBLK + blk] = M;
      psum[(size_t)(row0 + tid) * NBLK + blk] = S;
    }
  }
}

// --------------------- per-row log-sum-exp over blocks ---------------------
__global__ __launch_bounds__(256) void arc_rowreduce_kernel(
    const float* __restrict__ pmax, const float* __restrict__ psum,
    const float* __restrict__ loglab, float* __restrict__ nll) {
  __shared__ float sc[8];
  __shared__ float Msh;
  const int row = blockIdx.x;
  const float* pm = pmax + (size_t)row * NBLK;
  const float* ps = psum + (size_t)row * NBLK;
  float m = NEGBIG;
  for (int i = threadIdx.x; i < NBLK; i += 256) m = fmaxf(m, pm[i]);
  for (int off = 16; off > 0; off >>= 1) m = fmaxf(m, __shfl_xor(m, off, 32));
  if ((threadIdx.x & 31) == 0) sc[threadIdx.x >> 5] = m;
  __syncthreads();
  if (threadIdx.x == 0) {
    float M = sc[0];
    for (int w = 1; w < 8; ++w) M = fmaxf(M, sc[w]);
    Msh = M;
  }
  __syncthreads();
  const float M = Msh;
  float s = 0.f;
  for (int i = threadIdx.x; i < NBLK; i += 256) s += ps[i] * expf(pm[i] - M);
  for (int off = 16; off > 0; off >>= 1) s += __shfl_xor(s, off, 32);
  __syncthreads();
  if ((threadIdx.x & 31) == 0) sc[threadIdx.x >> 5] = s;
  __syncthreads();
  if (threadIdx.x == 0) {
    float S = sc[0] + sc[1] + sc[2] + sc[3] + sc[4] + sc[5] + sc[6] + sc[7];
    nll[row] = logf(S) + M - loglab[row];
  }
}

// ------------------------------- mean NLL ----------------------------------
__global__ __launch_bounds__(256) void arc_mean_kernel(
    const float* __restrict__ nll, float* __restrict__ out) {
  __shared__ float sc[8];
  float s = 0.f;
  for (int i = threadIdx.x; i < BB; i += 256) s += nll[i];
  for (int off = 16; off > 0; off >>= 1) s += __shfl_xor(s, off, 32);
  if ((threadIdx.x & 31) == 0) sc[threadIdx.x >> 5] = s;
  __syncthreads();
  if (threadIdx.x == 0) {
    float S = sc[0] + sc[1] + sc[2] + sc[3] + sc[4] + sc[5] + sc[6] + sc[7];
    out[0] = S / (float)BB;
  }
}

// ------------------------------- launcher ----------------------------------
extern "C" void kernel_launch(void* const* d_in, const int* in_sizes, int n_in,
                              void* d_out, int out_size, void* d_ws,
                              size_t ws_size, hipStream_t stream) {
  (void)in_sizes; (void)n_in; (void)out_size; (void)ws_size;
  const float* emb    = (const float*)d_in[0];  // (B, D) f32
  const int*   labels = (const int*)d_in[1];    // (B,) int
  const float* wgt    = (const float*)d_in[2];  // (C, D) f32
  float* out = (float*)d_out;

  float* ws     = (float*)d_ws;
  float* inv_e  = ws;                                  //  2048
  float* inv_w  = inv_e + BB;                          // 50000 (+pad)
  float* loglab = inv_w + 50048;                       //  2048
  float* pmax   = loglab + BB;                         //  2048*391
  float* psum   = pmax + (size_t)BB * NBLK;            //  2048*391
  float* nll    = psum + (size_t)BB * NBLK;            //  2048
  // total: ~6.4 MB of d_ws

  hipFuncSetAttribute(reinterpret_cast<const void*>(arc_main_kernel),
                      hipFuncAttributeMaxDynamicSharedMemorySize, SMEM_BYTES);

  arc_rownorm_kernel<<<BB, 128, 0, stream>>>(emb, inv_e);
  arc_rownorm_kernel<<<CC, 128, 0, stream>>>(wgt, inv_w);
  arc_main_kernel<<<NBLK, 256, SMEM_BYTES, stream>>>(
      emb, labels, wgt, inv_e, inv_w, pmax, psum, loglab);
  arc_rowreduce_kernel<<<BB, 256, 0, stream>>>(pmax, psum, loglab, nll);
  arc_mean_kernel<<<1, 256, 0, stream>>>(nll, out);
}